// GineBlock_67113158967915
// MI455X (gfx1250) — compile-verified
//
#include <hip/hip_runtime.h>

#define NN 100000
#define NE 1600000
#define DIM 128
#define BN_EPS 1e-5f

typedef float v2f __attribute__((ext_vector_type(2)));
typedef float v8f __attribute__((ext_vector_type(8)));

// ---------------------------------------------------------------------------
// Edge phase: msg = relu(x[src] + edge_attr); scatter-sum into aggr[dst].
// One wave32 per edge; each lane owns one float4 chunk (32*4 = 128 floats).
// ---------------------------------------------------------------------------
__global__ __launch_bounds__(256) void edge_scatter(
    const float* __restrict__ x, const int* __restrict__ ei,
    const float* __restrict__ ea, float* __restrict__ aggr)
{
    const int e    = (blockIdx.x * 256 + threadIdx.x) >> 5;   // edge id
    const int lane = threadIdx.x & 31;
    if (e >= NE) return;

    const int src = ei[e];
    const int dst = ei[NE + e];

    const float4 xv = ((const float4*)(x  + (size_t)src * DIM))[lane];
    const float4 ev = ((const float4*)(ea + (size_t)e   * DIM))[lane];

    float4 m;
    m.x = fmaxf(xv.x + ev.x, 0.f);
    m.y = fmaxf(xv.y + ev.y, 0.f);
    m.z = fmaxf(xv.z + ev.z, 0.f);
    m.w = fmaxf(xv.w + ev.w, 0.f);

    float* ap = aggr + (size_t)dst * DIM + lane * 4;
    unsafeAtomicAdd(ap + 0, m.x);
    unsafeAtomicAdd(ap + 1, m.y);
    unsafeAtomicAdd(ap + 2, m.z);
    unsafeAtomicAdd(ap + 3, m.w);
}

// ---------------------------------------------------------------------------
// GEMM: out[n,j] = act( sum_k in[n,k] * W[j,k] + bias[j] )
//   in = A0 (+ A1 when ADD2)  -- fuses h0 = x + aggr for layer 1
// Block = 256 threads = 8 waves. Each wave: 16 rows x 128 cols using
// V_WMMA_F32_16X16X4_F32, K swept 4 at a time (32 steps), 8 column tiles.
// W (128x128 row-major, W[j][k]) staged in 64KB LDS.
// ---------------------------------------------------------------------------
template <bool ADD2, bool RELU>
__global__ __launch_bounds__(256) void gemm_wmma(
    const float* __restrict__ A0, const float* __restrict__ A1,
    const float* __restrict__ W,  const float* __restrict__ bias,
    float* __restrict__ out, int nrows)
{
    __shared__ float lw[DIM * DIM];   // 64 KB

    // cooperative load of W into LDS (float4 per thread iteration)
    for (int i = threadIdx.x; i < DIM * DIM / 4; i += 256) {
        ((float4*)lw)[i] = ((const float4*)W)[i];
    }
    __syncthreads();

    const int waveId = threadIdx.x >> 5;
    const int lane   = threadIdx.x & 31;
    const int laneLo = lane & 15;
    const int laneHi = lane >> 4;

    const int rowBase = blockIdx.x * 128 + waveId * 16;
    int arow = rowBase + laneLo;
    if (arow >= nrows) arow = 0;              // clamp loads; stores are guarded

    const float* __restrict__ ap0 = A0 + (size_t)arow * DIM;
    const float* __restrict__ ap1 = ADD2 ? (A1 + (size_t)arow * DIM) : nullptr;

    v8f acc[8] = {};                           // 8 column tiles of 16

    for (int kb = 0; kb < 32; ++kb) {
        const int koff = kb * 4 + laneHi * 2;  // this lane's K pair
        v2f afrag = *(const v2f*)(ap0 + koff); // A[m= lane%16][koff, koff+1]
        if (ADD2) {
            v2f a2 = *(const v2f*)(ap1 + koff);
            afrag.x += a2.x;
            afrag.y += a2.y;
        }
#pragma unroll
        for (int t = 0; t < 8; ++t) {
            // B[k][n] = W[n][k]; lane holds N = lane%16, K pair = koff..koff+1
            v2f bfrag = *(const v2f*)(lw + (t * 16 + laneLo) * DIM + koff);
            acc[t] = __builtin_amdgcn_wmma_f32_16x16x4_f32(
                false, afrag, false, bfrag, (short)0, acc[t], false, false);
        }
    }

    // D layout: VGPR r, lanes 0-15 -> M=r, lanes 16-31 -> M=8+r; N = lane%16
#pragma unroll
    for (int t = 0; t < 8; ++t) {
        const int col = t * 16 + laneLo;
        const float bv = bias[col];
#pragma unroll
        for (int r = 0; r < 8; ++r) {
            const int row = rowBase + laneHi * 8 + r;
            if (row < nrows) {
                float v = acc[t][r] + bv;
                if (RELU) v = fmaxf(v, 0.f);
                out[(size_t)row * DIM + col] = v;
            }
        }
    }
}

// ---------------------------------------------------------------------------
// BatchNorm statistics: per-column sum and sum of squares into sums[0..127]
// and sums[128..255]. 2 threads per column per block, 512 rows per block.
// ---------------------------------------------------------------------------
#define BN_ROWS 512
__global__ __launch_bounds__(256) void bn_stats(
    const float* __restrict__ h, float* __restrict__ sums)
{
    __shared__ float rs[256];
    __shared__ float rq[256];

    const int col  = threadIdx.x & 127;
    const int half = threadIdx.x >> 7;            // 0 or 1
    const int rEnd = min(NN, (int)(blockIdx.x + 1) * BN_ROWS);

    float s = 0.f, q = 0.f;
    for (int r = blockIdx.x * BN_ROWS + half; r < rEnd; r += 2) {
        const float v = h[(size_t)r * DIM + col];
        s += v;
        q += v * v;
    }
    rs[threadIdx.x] = s;
    rq[threadIdx.x] = q;
    __syncthreads();
    if (threadIdx.x < 128) {
        s = rs[threadIdx.x] + rs[threadIdx.x + 128];
        q = rq[threadIdx.x] + rq[threadIdx.x + 128];
        unsafeAtomicAdd(&sums[col], s);
        unsafeAtomicAdd(&sums[DIM + col], q);
    }
}

// ---------------------------------------------------------------------------
// BatchNorm apply + final ReLU: out = relu((h-mean)*rsqrt(var+eps)*gamma+beta)
// float4 vectorized; mean/var recomputed per thread from the column sums.
// ---------------------------------------------------------------------------
__global__ __launch_bounds__(256) void bn_apply(
    const float* __restrict__ h, const float* __restrict__ sums,
    const float* __restrict__ gamma, const float* __restrict__ beta,
    float* __restrict__ out)
{
    const size_t i = (size_t)blockIdx.x * 256 + threadIdx.x; // float4 index
    if (i >= (size_t)NN * DIM / 4) return;

    const int c0 = ((int)(i & 31)) * 4;      // 32 float4 per row
    const float4 hv = ((const float4*)h)[i];
    const float invN = 1.0f / (float)NN;

    float r[4] = {hv.x, hv.y, hv.z, hv.w};
#pragma unroll
    for (int j = 0; j < 4; ++j) {
        const int c = c0 + j;
        const float mean = sums[c] * invN;
        const float var  = sums[DIM + c] * invN - mean * mean;
        const float inv  = rsqrtf(var + BN_EPS);
        r[j] = fmaxf((r[j] - mean) * inv * gamma[c] + beta[c], 0.f);
    }
    float4 o = {r[0], r[1], r[2], r[3]};
    ((float4*)out)[i] = o;
}

// ---------------------------------------------------------------------------
extern "C" void kernel_launch(void* const* d_in, const int* in_sizes, int n_in,
                              void* d_out, int out_size, void* d_ws, size_t ws_size,
                              hipStream_t stream) {
    const float* x     = (const float*)d_in[0];
    const int*   ei    = (const int*)d_in[1];   // [2, NE] row-major
    const float* ea    = (const float*)d_in[2];
    const float* W1    = (const float*)d_in[3];
    const float* b1    = (const float*)d_in[4];
    const float* W2    = (const float*)d_in[5];
    const float* b2    = (const float*)d_in[6];
    const float* gamma = (const float*)d_in[7];
    const float* beta  = (const float*)d_in[8];

    float* buf0 = (float*)d_ws;                   // aggr, then reused for h
    float* buf1 = buf0 + (size_t)NN * DIM;        // t = relu(h0 W1^T + b1)
    float* sums = buf1 + (size_t)NN * DIM;        // 256 floats: sum | sumsq

    hipMemsetAsync(buf0, 0, (size_t)NN * DIM * sizeof(float), stream);
    hipMemsetAsync(sums, 0, 2 * DIM * sizeof(float), stream);

    edge_scatter<<<NE / 8, 256, 0, stream>>>(x, ei, ea, buf0);

    const int gemmGrid = (NN + 127) / 128;
    gemm_wmma<true,  true ><<<gemmGrid, 256, 0, stream>>>(x,    buf0, W1, b1, buf1, NN);
    gemm_wmma<false, false><<<gemmGrid, 256, 0, stream>>>(buf1, nullptr, W2, b2, buf0, NN);

    bn_stats<<<(NN + BN_ROWS - 1) / BN_ROWS, 256, 0, stream>>>(buf0, sums);

    const size_t nvec = (size_t)NN * DIM / 4;
    bn_apply<<<(int)((nvec + 255) / 256), 256, 0, stream>>>(buf0, sums, gamma, beta,
                                                            (float*)d_out);
}